// RotatedIoULoss_54468775248534
// MI455X (gfx1250) — compile-verified
//
#include <hip/hip_runtime.h>
#include <math.h>

// ---------------------------------------------------------------------------
// RotatedIoULoss for MI455X (gfx1250).
//
// 44 B/pair vs ~1.5-2k VALU ops/pair -> heavily compute bound over the
// 23.3 TB/s HBM floor. No shared-K contraction exists, so WMMA is not
// applicable; the win is minimizing VALU work: pseudo-angle keys (no atan2),
// fast divides, a key-only min/max Batcher network (index embedded in low
// mantissa bits), and an LDS-gather for the sorted points (bank-conflict-free
// [k][tid] layout). Inputs are fetched via the gfx1250 async Global->LDS
// path (ASYNCcnt) with speculative prefetch.
// ---------------------------------------------------------------------------

#ifndef __has_builtin
#define __has_builtin(x) 0
#endif

#define BLK 256
#define EPS_NUM 1e-8f
#define BIGV 1000000.0f
#define NVERT 24

#if defined(__HIP_DEVICE_COMPILE__) && __has_builtin(__builtin_amdgcn_global_load_async_to_lds_b32)
#define USE_ASYNC_LDS 1
#else
#define USE_ASYNC_LDS 0
#endif

#if USE_ASYNC_LDS
// Signature per hipcc: (AS(1) int* src, AS(3) int* dst, imm offset, imm cpol).
typedef __attribute__((address_space(1))) int* as1_iptr;
typedef __attribute__((address_space(3))) int* as3_iptr;
#endif

// Monotone surrogate for atan2: strictly increasing map of angle onto (-2, 2].
__device__ __forceinline__ float pseudo_angle(float x, float y) {
    float d = fabsf(x) + fabsf(y);
    float p = copysignf(1.0f - __fdividef(x, d), y);
    return (d > 0.0f) ? p : 0.0f;  // atan2(0,0) == 0
}

// Key-only comparator: values are unique (index in low mantissa bits), so
// (min,max) is an exact compare-exchange. 2 full-rate VALU ops, no v_cmp.
__device__ __forceinline__ void cswap1(float& a, float& b) {
    const float lo = fminf(a, b);
    const float hi = fmaxf(a, b);
    a = lo; b = hi;
}

__global__ __launch_bounds__(BLK) void rotated_iou_loss_kernel(
    const float* __restrict__ pred, const float* __restrict__ target,
    float* __restrict__ out, int N) {
    const int i = blockIdx.x * BLK + threadIdx.x;
    const int tid = threadIdx.x;
    const int ii = (i < N) ? i : (N - 1);  // clamp so every lane has a valid addr

    // Gather buffer: layout [k][tid] (row stride 256 float2 = 512 dwords ==
    // 0 mod 64 banks) -> stores and idx-gathers are both bank-conflict free.
    // Each thread touches only column `tid`, so no barrier is required.
    __shared__ float2 gbuf[NVERT * BLK];

    // Warm cache ahead (speculative prefetch, safe past end of array).
    __builtin_prefetch(pred + (size_t)ii * 5 + 5 * BLK, 0, 3);
    __builtin_prefetch(target + (size_t)ii * 5 + 5 * BLK, 0, 3);

    float b1[5], b2[5];
#if USE_ASYNC_LDS
    // Stage inputs Global->LDS with the gfx1250 async copy path (ASYNCcnt).
    // Stride-5 dword layout: gcd(5,64)=1 -> bank-conflict free.
    __shared__ float sP[BLK * 5];
    __shared__ float sT[BLK * 5];
    {
        const float* gp = pred + (size_t)ii * 5;
        const float* gt = target + (size_t)ii * 5;
        float* lp = &sP[tid * 5];
        float* lt = &sT[tid * 5];
#pragma unroll
        for (int k = 0; k < 5; ++k) {
            __builtin_amdgcn_global_load_async_to_lds_b32(
                (as1_iptr)(unsigned long long)(gp + k),
                (as3_iptr)(unsigned int)(unsigned long long)(lp + k), 0, 0);
            __builtin_amdgcn_global_load_async_to_lds_b32(
                (as1_iptr)(unsigned long long)(gt + k),
                (as3_iptr)(unsigned int)(unsigned long long)(lt + k), 0, 0);
        }
#if __has_builtin(__builtin_amdgcn_s_wait_asynccnt)
        __builtin_amdgcn_s_wait_asynccnt(0);
#else
        asm volatile("s_wait_asynccnt 0" ::: "memory");
#endif
        asm volatile("" ::: "memory");  // keep ds reads below the wait
#pragma unroll
        for (int k = 0; k < 5; ++k) { b1[k] = lp[k]; b2[k] = lt[k]; }
    }
#else
    {
        const float* gp = pred + (size_t)ii * 5;
        const float* gt = target + (size_t)ii * 5;
#pragma unroll
        for (int k = 0; k < 5; ++k) { b1[k] = gp[k]; b2[k] = gt[k]; }
    }
#endif
    if (i >= N) return;

    // ---- corners ----------------------------------------------------------
    float c1x[4], c1y[4], c2x[4], c2y[4];
    {
        float s, c;
        __sincosf(b1[4], &s, &c);
        const float hw = 0.5f * b1[2], hh = 0.5f * b1[3];
        const float xs[4] = {hw, -hw, -hw, hw};
        const float ys[4] = {hh, hh, -hh, -hh};
#pragma unroll
        for (int k = 0; k < 4; ++k) {
            c1x[k] = xs[k] * c - ys[k] * s + b1[0];
            c1y[k] = xs[k] * s + ys[k] * c + b1[1];
        }
    }
    {
        float s, c;
        __sincosf(b2[4], &s, &c);
        const float hw = 0.5f * b2[2], hh = 0.5f * b2[3];
        const float xs[4] = {hw, -hw, -hw, hw};
        const float ys[4] = {hh, hh, -hh, -hh};
#pragma unroll
        for (int k = 0; k < 4; ++k) {
            c2x[k] = xs[k] * c - ys[k] * s + b2[0];
            c2y[k] = xs[k] * s + ys[k] * c + b2[1];
        }
    }

    // ---- candidate vertices: 4 + 4 corners, 16 edge intersections ---------
    float vx[NVERT], vy[NVERT];
    bool msk[NVERT];
#pragma unroll
    for (int k = 0; k < 4; ++k) { vx[k] = c1x[k]; vy[k] = c1y[k]; }
#pragma unroll
    for (int k = 0; k < 4; ++k) { vx[4 + k] = c2x[k]; vy[4 + k] = c2y[k]; }

#pragma unroll
    for (int a = 0; a < 4; ++a) {
        const float x1 = c1x[a], y1 = c1y[a];
        const float ex = c1x[(a + 1) & 3] - x1, ey = c1y[(a + 1) & 3] - y1;
#pragma unroll
        for (int b = 0; b < 4; ++b) {
            const float x3 = c2x[b], y3 = c2y[b];
            const float gx = y3 - c2y[(b + 1) & 3];   // (y3 - y4)
            const float hx = x3 - c2x[(b + 1) & 3];   // (x3 - x4)
            const float num = ex * gx - ey * hx;
            const float den_t = (x1 - x3) * gx - (y1 - y3) * hx;
            const float den_u = ex * (y1 - y3) - ey * (x1 - x3);
            const bool par = (num == 0.0f);
            const float rn = __fdividef(1.0f, par ? 1.0f : num);
            const float t = par ? -1.0f : den_t * rn;
            const float u = par ? -1.0f : den_u * rn;
            const bool m = (t > 0.0f) & (t < 1.0f) & (u > 0.0f) & (u < 1.0f);
            const float t2 = __fdividef(den_t, num + EPS_NUM);
            const int s = 8 + a * 4 + b;
            vx[s] = m ? (x1 + t2 * ex) : 0.0f;
            vy[s] = m ? (y1 + t2 * ey) : 0.0f;
            msk[s] = m;
        }
    }

    // ---- corner-in-box tests ----------------------------------------------
    {  // c1 corners inside box2
        const float ax = c2x[0], ay = c2y[0];
        const float abx = c2x[1] - ax, aby = c2y[1] - ay;
        const float adx = c2x[3] - ax, ady = c2y[3] - ay;
        const float iab = __fdividef(1.0f, abx * abx + aby * aby);
        const float iad = __fdividef(1.0f, adx * adx + ady * ady);
#pragma unroll
        for (int k = 0; k < 4; ++k) {
            const float amx = c1x[k] - ax, amy = c1y[k] - ay;
            const float rab = (abx * amx + aby * amy) * iab;
            const float rad = (adx * amx + ady * amy) * iad;
            msk[k] = (rab > -1e-6f) & (rab < 1.0f + 1e-6f) &
                     (rad > -1e-6f) & (rad < 1.0f + 1e-6f);
        }
    }
    {  // c2 corners inside box1
        const float ax = c1x[0], ay = c1y[0];
        const float abx = c1x[1] - ax, aby = c1y[1] - ay;
        const float adx = c1x[3] - ax, ady = c1y[3] - ay;
        const float iab = __fdividef(1.0f, abx * abx + aby * aby);
        const float iad = __fdividef(1.0f, adx * adx + ady * ady);
#pragma unroll
        for (int k = 0; k < 4; ++k) {
            const float amx = c2x[k] - ax, amy = c2y[k] - ay;
            const float rab = (abx * amx + aby * amy) * iab;
            const float rad = (adx * amx + ady * amy) * iad;
            msk[4 + k] = (rab > -1e-6f) & (rab < 1.0f + 1e-6f) &
                         (rad > -1e-6f) & (rad < 1.0f + 1e-6f);
        }
    }

    // ---- masked mean ------------------------------------------------------
    float sx = 0.0f, sy = 0.0f;
    int cnt = 0;
#pragma unroll
    for (int k = 0; k < NVERT; ++k) {
        sx += msk[k] ? vx[k] : 0.0f;
        sy += msk[k] ? vy[k] : 0.0f;
        cnt += msk[k] ? 1 : 0;
    }
    const float invn = __fdividef(1.0f, (float)((cnt < 1) ? 1 : cnt));
    const float mx = sx * invn, my = sy * invn;

    // ---- center, park points in LDS, build index-embedded sort keys --------
    float ang[NVERT];
#pragma unroll
    for (int k = 0; k < NVERT; ++k) {
        const float dx = vx[k] - mx, dy = vy[k] - my;
        gbuf[(k << 8) + tid] = make_float2(dx, dy);  // ds_store_b64, offset k*2048
        const float key = msk[k] ? pseudo_angle(dx, dy) : BIGV;
        // Unique keys: low 5 mantissa bits carry the candidate index.
        ang[k] = __uint_as_float((__float_as_uint(key) & ~31u) | (unsigned)k);
    }

    // ---- Batcher odd-even mergesort on keys only (min/max comparators) -----
#pragma unroll
    for (int p = 1; p < NVERT; p <<= 1) {
#pragma unroll
        for (int k = p; k > 0; k >>= 1) {
#pragma unroll
            for (int j = k % p; j + k < NVERT; j += 2 * k) {
#pragma unroll
                for (int q = 0; q < k; ++q) {
                    const int a = j + q, b = j + q + k;
                    if (b < NVERT && (a / (2 * p)) == (b / (2 * p)))
                        cswap1(ang[a], ang[b]);
                }
            }
        }
    }

    // ---- gather in sorted order + invalid-replacement + shoelace, fused ----
    float fxp = 0.0f, fyp = 0.0f, prx = 0.0f, pry = 0.0f, area2 = 0.0f;
#pragma unroll
    for (int k = 0; k < NVERT; ++k) {
        const int idx = (int)(__float_as_uint(ang[k]) & 31u);
        const float2 p = gbuf[(idx << 8) + tid];  // conflict-free gather
        if (k == 0) {
            fxp = p.x; fyp = p.y; prx = p.x; pry = p.y;
        } else {
            const bool inv = ang[k] > 3.0f;  // valid keys lie in [-2, 2]
            const float px = inv ? fxp : p.x;
            const float py = inv ? fyp : p.y;
            area2 += prx * py - pry * px;
            prx = px; pry = py;
        }
    }
    area2 += prx * fyp - pry * fxp;  // close the polygon
    const float area_i = 0.5f * fabsf(area2);

    // ---- IoU loss ----------------------------------------------------------
    const float a1 = b1[2] * b1[3];
    const float a2 = b2[2] * b2[3];
    const float iou = __fdividef(area_i, a1 + a2 - area_i);
    out[i] = 1.0f - fmaxf(iou, 0.1f);
}

extern "C" void kernel_launch(void* const* d_in, const int* in_sizes, int n_in,
                              void* d_out, int out_size, void* d_ws, size_t ws_size,
                              hipStream_t stream) {
    (void)n_in; (void)d_ws; (void)ws_size;
    const float* pred = (const float*)d_in[0];
    const float* target = (const float*)d_in[1];
    float* out = (float*)d_out;
    const int N = in_sizes[0] / 5;
    if (N <= 0) return;
    (void)out_size;
    const int grid = (N + BLK - 1) / BLK;
    rotated_iou_loss_kernel<<<grid, BLK, 0, stream>>>(pred, target, out, N);
}